// FratPirorBlock_25812753448988
// MI455X (gfx1250) — compile-verified
//
#include <hip/hip_runtime.h>
#include <cstdint>
#include <cmath>

// Problem constants (from the reference)
#define KDIM   224               // input spatial size
#define SWIN   3                 // unfold window
#define SUMD   222               // KDIM - SWIN + 1
#define PLANE  (KDIM * KDIM)     // 50176 floats per (b,c) plane = 200704 B
#define NP     (SWIN * SWIN)     // 9 patch offsets
#define GROUPS ((SUMD * SUMD) / NP)  // 5476 groups of 9 per patch offset (exact)
#define THREADS 512              // 16 wave32s per workgroup

// ---- CDNA5 async global->LDS copy (ASYNCcnt path) -------------------------
// Per-lane: copies 16 bytes from the global address in gaddr to the LDS offset in lds_off. Tracked by ASYNCcnt.
__device__ __forceinline__ void async_ld_b128(uint32_t lds_off, const float* gaddr) {
    asm volatile("global_load_async_to_lds_b128 %0, %1, off"
                 :: "v"(lds_off), "v"(gaddr)
                 : "memory");
}

__device__ __forceinline__ void wait_asynccnt0() {
    asm volatile("s_wait_asynccnt 0x0" ::: "memory");
}

// ---------------------------------------------------------------------------
// One workgroup per (b,c) plane. Stage 224x224 f32 plane (200704 B) into the
// 320KB CDNA5 LDS via async b128 copies, then compute the 9 patch-offset
// passes entirely out of LDS (each HBM byte is read exactly once).
// ---------------------------------------------------------------------------
extern __shared__ float lds[];   // dynamic: PLANE * 4 bytes

__global__ void __launch_bounds__(THREADS)
frat_range_kernel(const float* __restrict__ x, float* __restrict__ out,
                  int nplanes, int out_size) {
    const int plane = blockIdx.x;
    if (plane >= nplanes) return;

    const float*   src      = x + (size_t)plane * PLANE;
    const int      tid      = (int)threadIdx.x;
    // Low 32 bits of a generic pointer to LDS == byte offset within the
    // workgroup's LDS allocation (what the async-load VDST VGPR expects).
    const uint32_t lds_base = (uint32_t)(uintptr_t)(const void*)lds;

    // ---- Stage the plane: 12544 x b128 lane-copies, 512B per wave-inst ----
    for (int k = tid; k < PLANE / 4; k += THREADS) {
        async_ld_b128(lds_base + 16u * (uint32_t)k, src + 4 * k);
    }
    wait_asynccnt0();     // my wave's async copies landed in LDS
    __syncthreads();      // everyone's copies visible to everyone

    // ---- 9 passes over LDS: disjoint 9-element windows of the flattened
    //      222x222 sub-image at offset (i,j). Stride-9 dwords across lanes
    //      (9 coprime with 64 banks -> conflict-free).
    float acc = 0.0f;
#pragma unroll
    for (int p = 0; p < NP; ++p) {
        const int i = p / SWIN;
        const int j = p - i * SWIN;
        const float* base = lds + i * KDIM + j;
        for (int m = tid; m < GROUPS; m += THREADS) {
            int s   = 9 * m;                 // flat index in 222x222 sub-image
            int u   = s / SUMD;
            int v   = s - u * SUMD;
            int idx = u * KDIM + v;          // index into 224-stride LDS plane
            float mn = INFINITY, mx = -INFINITY;
#pragma unroll
            for (int r = 0; r < 9; ++r) {
                float val = base[idx];
                mn = fminf(mn, val);
                mx = fmaxf(mx, val);
                ++v;
                if (v == SUMD) { v = 0; idx += 3; }   // (u,221)->(u+1,0): +3
                else           { ++idx; }
            }
            acc += (mx - mn);
        }
    }

    // ---- Workgroup reduction (plane data no longer needed; reuse LDS) ----
    __syncthreads();
    lds[tid] = acc;
    __syncthreads();
    for (int off = THREADS / 2; off > 0; off >>= 1) {
        if (tid < off) lds[tid] += lds[tid + off];
        __syncthreads();
    }

    if (tid == 0) {
        // tp = sum_q( range_q * (K/S) + 1 ) = (224/3)*sum(range) + 49284
        float tp = lds[0] * (224.0f / 3.0f) + (float)(NP * GROUPS);
        out[plane] = logf(tp);
        if (plane == 0 && out_size > nplanes) {
            out[nplanes] = logf(224.0f / 3.0f);   // trailing scalar log(K/S)
        }
    }
}

// ---------------------------------------------------------------------------
extern "C" void kernel_launch(void* const* d_in, const int* in_sizes, int n_in,
                              void* d_out, int out_size, void* d_ws, size_t ws_size,
                              hipStream_t stream) {
    (void)n_in; (void)d_ws; (void)ws_size;
    const float* x   = (const float*)d_in[0];
    float*       out = (float*)d_out;

    const int nplanes = in_sizes[0] / PLANE;        // 16 * 64 = 1024
    const size_t shmem = (size_t)PLANE * sizeof(float);  // 200704 B (< 320 KB WG LDS)

    // Allow >64KB dynamic LDS (CDNA5 workgroups may allocate up to 320 KB).
    (void)hipFuncSetAttribute((const void*)frat_range_kernel,
                              hipFuncAttributeMaxDynamicSharedMemorySize,
                              (int)shmem);

    frat_range_kernel<<<dim3(nplanes), dim3(THREADS), shmem, stream>>>(
        x, out, nplanes, out_size);
}